// GraphAttentionLayer_4956392259899
// MI455X (gfx1250) — compile-verified
//
#include <hip/hip_runtime.h>

#define NNODES   100000
#define NEDGES   1600000
#define IN_DIM   256
#define OUT_DIM  64
#define NEG_SLOPE 0.2f
#define EPS_F     1e-9f

typedef __attribute__((ext_vector_type(2))) float v2f;
typedef __attribute__((ext_vector_type(8))) float v8f;

// ---------------------------------------------------------------------------
// Kernel 1: h = X @ W^T  via V_WMMA_F32_16X16X4_F32
// block = 256 threads (8 waves). Each wave computes a 16-row strip of h
// across all 64 output columns (4 tiles of N=16), K swept 256 in steps of 4.
//
// W staged in LDS pair-interleaved so each lane's B fragment is a contiguous,
// 8B-aligned v2f (single ds_load_b64 straight into an even VGPR pair):
//   wlds[(kk>>1)*128 + col*2 + (kk&1)]  ==  W^T[kk][col]      (64 KB total)
// B fragment for K-step kb, tile t:  *(v2f*)&wlds[(kb+kh)*64 + 2*col]
//   (kh = 0 for lanes 0-15, 2 for lanes 16-31, per ISA 7.12.2 f32 layouts)
// A fragment: lane L holds X[strip + (L&15)][kb + kh + {0,1}] -> global b64.
// ---------------------------------------------------------------------------
__global__ __launch_bounds__(256)
void gat_gemm_h(const float* __restrict__ X, const float* __restrict__ W,
                float* __restrict__ h) {
  __shared__ float wlds[IN_DIM * OUT_DIM];   // 64 KB
  const int tid = threadIdx.x;
  for (int i = tid; i < IN_DIM * OUT_DIM; i += 256) {
    int col = i >> 8;       // 0..63  (W is [64][256] row-major)
    int kk  = i & 255;      // 0..255
    wlds[(kk >> 1) * 128 + col * 2 + (kk & 1)] = W[i];
  }
  __syncthreads();

  const int lane = tid & 31;
  const int wave = tid >> 5;
  const int m    = lane & 15;           // M (row in strip) / N (col in tile)
  const int kh   = (lane >> 4) << 1;    // K sub-offset: 0 or 2
  int row  = blockIdx.x * 128 + wave * 16 + m;
  int lrow = row < NNODES ? row : NNODES - 1;   // clamp tail loads
  const float* xrow = X + (size_t)lrow * IN_DIM + kh;

  v8f acc[4] = {};
#pragma unroll 1
  for (int kb0 = 0; kb0 < IN_DIM; kb0 += 16) {
    // pull the next 64B of this row ahead of the load stream
    __builtin_prefetch(xrow + kb0 + 16, 0, 0);   // global_prefetch_b8
#pragma unroll
    for (int kb = kb0; kb < kb0 + 16; kb += 4) {
      v2f a = *(const v2f*)(xrow + kb);                    // global_load_b64
      const float* bbase = wlds + (kb + kh) * 64 + 2 * m;  // tile 0 fragment
#pragma unroll
      for (int t = 0; t < 4; ++t) {
        v2f b = *(const v2f*)(bbase + t * 32);             // ds_load_b64
        acc[t] = __builtin_amdgcn_wmma_f32_16x16x4_f32(
            /*neg_a=*/false, a, /*neg_b=*/false, b,
            /*c_mod=*/(short)0, acc[t], /*reuse_a=*/false, /*reuse_b=*/false);
      }
    }
  }

  // C/D layout: VGPR r -> M=r (lanes 0-15), M=r+8 (lanes 16-31); N = lane&15
  const int rbase = blockIdx.x * 128 + wave * 16 + ((lane < 16) ? 0 : 8);
  const bool full = (blockIdx.x + 1) * 128 <= NNODES;   // uniform fast path
  if (full) {
#pragma unroll
    for (int t = 0; t < 4; ++t)
#pragma unroll
      for (int r = 0; r < 8; ++r)
        h[(size_t)(rbase + r) * OUT_DIM + t * 16 + m] = acc[t][r];
  } else {
#pragma unroll
    for (int t = 0; t < 4; ++t)
#pragma unroll
      for (int r = 0; r < 8; ++r)
        if (rbase + r < NNODES)
          h[(size_t)(rbase + r) * OUT_DIM + t * 16 + m] = acc[t][r];
  }
}

// ---------------------------------------------------------------------------
// Kernel 2: per-node scores a_src = h.attn[:64], a_dst = h.attn[64:]
// Also initializes segment-max keys (encoded -inf) and segment sums.
// h is L2-resident (25.6 MB << 192 MB) right after kernel 1.
// ---------------------------------------------------------------------------
__global__ __launch_bounds__(256)
void gat_node_scores(const float* __restrict__ h, const float* __restrict__ attn,
                     float* __restrict__ a_src, float* __restrict__ a_dst,
                     unsigned* __restrict__ maxkey, float* __restrict__ sumexp) {
  int n = blockIdx.x * 256 + threadIdx.x;
  if (n >= NNODES) return;
  const float* hr = h + (size_t)n * OUT_DIM;
  float s = 0.f, d = 0.f;
#pragma unroll
  for (int i = 0; i < OUT_DIM; ++i) {
    float hv = hr[i];
    s += hv * attn[i];
    d += hv * attn[OUT_DIM + i];
  }
  a_src[n]  = s;
  a_dst[n]  = d;
  maxkey[n] = 0x007FFFFFu;   // monotonic-uint encoding of -inf
  sumexp[n] = 0.f;
}

// out[n,:] = bias (must pre-fill before atomic scatter)
__global__ void gat_init_out(float* __restrict__ out, const float* __restrict__ bias) {
  int i = blockIdx.x * 256 + threadIdx.x;
  if (i < NNODES * OUT_DIM) out[i] = bias[i & (OUT_DIM - 1)];
}

// ---------------------------------------------------------------------------
// Kernel 3: e = leaky_relu(a_src[src] + a_dst[dst]); segment max over dst
// via monotonic uint encoding + atomicMax (global_atomic_max_u32, L2-served).
// ---------------------------------------------------------------------------
__global__ void gat_edge_max(const int* __restrict__ ei,
                             const float* __restrict__ a_src,
                             const float* __restrict__ a_dst,
                             float* __restrict__ ebuf,
                             unsigned* __restrict__ maxkey) {
  int e = blockIdx.x * 256 + threadIdx.x;
  if (e >= NEDGES) return;
  int s = ei[e], d = ei[NEDGES + e];
  float v = a_src[s] + a_dst[d];
  v = (v >= 0.f) ? v : NEG_SLOPE * v;
  ebuf[e] = v;
  unsigned u   = __float_as_uint(v);
  unsigned key = (u >> 31) ? ~u : (u | 0x80000000u);
  atomicMax(maxkey + d, key);
}

// Kernel 4: exp(e - max[dst]) and segment sum (global_atomic_add_f32)
__global__ void gat_edge_expsum(const int* __restrict__ ei,
                                const unsigned* __restrict__ maxkey,
                                float* __restrict__ ebuf,
                                float* __restrict__ sumexp) {
  int e = blockIdx.x * 256 + threadIdx.x;
  if (e >= NEDGES) return;
  int d = ei[NEDGES + e];
  unsigned key = maxkey[d];
  unsigned u   = (key >> 31) ? (key & 0x7fffffffu) : ~key;
  float ex = __expf(ebuf[e] - __uint_as_float(u));
  ebuf[e] = ex;
  atomicAdd(sumexp + d, ex);
}

// ---------------------------------------------------------------------------
// Kernel 5: out[dst,:] += (exp/(sum+eps)) * h[src,:]
// One wave per edge: lanes cover the 64-wide row as two coalesced 128B
// bursts of global_atomic_add_f32. h and out both fit in L2.
// ---------------------------------------------------------------------------
__global__ __launch_bounds__(256)
void gat_edge_scatter(const int* __restrict__ ei,
                      const float* __restrict__ ebuf,
                      const float* __restrict__ sumexp,
                      const float* __restrict__ h,
                      float* __restrict__ out) {
  int e    = blockIdx.x * 8 + (threadIdx.x >> 5);
  int lane = threadIdx.x & 31;
  if (e >= NEDGES) return;
  int s = ei[e], d = ei[NEDGES + e];
  float att = ebuf[e] / (sumexp[d] + EPS_F);
  float h0 = h[(size_t)s * OUT_DIM + lane];
  float h1 = h[(size_t)s * OUT_DIM + 32 + lane];
  atomicAdd(out + (size_t)d * OUT_DIM + lane,      att * h0);
  atomicAdd(out + (size_t)d * OUT_DIM + 32 + lane, att * h1);
}

// ---------------------------------------------------------------------------
extern "C" void kernel_launch(void* const* d_in, const int* in_sizes, int n_in,
                              void* d_out, int out_size, void* d_ws, size_t ws_size,
                              hipStream_t stream) {
  const float* X    = (const float*)d_in[0];   // [N,256]
  const int*   ei   = (const int*)  d_in[1];   // [2,E]
  const float* W    = (const float*)d_in[2];   // [64,256]
  const float* attn = (const float*)d_in[3];   // [128]
  const float* bias = (const float*)d_in[4];   // [64]
  float* out = (float*)d_out;                  // [N,64]

  // workspace layout (bytes): total ~33.6 MB
  char* ws = (char*)d_ws;
  float*    h      = (float*)   (ws);               // N*64 f32 = 25.6 MB
  float*    a_src  = (float*)   (ws + 25600000);    // N f32
  float*    a_dst  = (float*)   (ws + 26000000);    // N f32
  unsigned* maxkey = (unsigned*)(ws + 26400000);    // N u32
  float*    sumexp = (float*)   (ws + 26800000);    // N f32
  float*    ebuf   = (float*)   (ws + 27200000);    // E f32 = 6.4 MB

  gat_gemm_h<<<(NNODES + 127) / 128, 256, 0, stream>>>(X, W, h);
  gat_node_scores<<<(NNODES + 255) / 256, 256, 0, stream>>>(h, attn, a_src, a_dst,
                                                            maxkey, sumexp);
  gat_init_out<<<(NNODES * OUT_DIM + 255) / 256, 256, 0, stream>>>(out, bias);
  gat_edge_max<<<(NEDGES + 255) / 256, 256, 0, stream>>>(ei, a_src, a_dst, ebuf, maxkey);
  gat_edge_expsum<<<(NEDGES + 255) / 256, 256, 0, stream>>>(ei, maxkey, ebuf, sumexp);
  gat_edge_scatter<<<(NEDGES * 32 + 255) / 256, 256, 0, stream>>>(ei, ebuf, sumexp, h, out);
}